// ProjectedGaussianRasterizer_7421703487874
// MI455X (gfx1250) — compile-verified
//
#include <hip/hip_runtime.h>
#include <hip/hip_bf16.h>

#define G_N 1024
#define IMG_W 192
#define IMG_H 192
#define NPIX (IMG_W * IMG_H)
#define FRONT_K 8
#define ALPHA_THRESH (1.0f / 255.0f)
#define ALPHA_CLAMP 0.999f
#define CSHIFT 96.0f   // center-shift coords to reduce f32 cancellation
#define SIG_STRIDE 20  // padded row stride (floats): 16B-aligned, conflict-free

typedef __attribute__((ext_vector_type(2))) float v2f;
typedef __attribute__((ext_vector_type(8))) float v8f;

// ---------------------------------------------------------------------------
// Kernel 1: stable depth sort (rank via O(G^2) scan over LDS depths) +
// build the per-gaussian rank-6 feature vector for the WMMA bilinear form,
// packed directly in V_WMMA_F32_16X16X4_F32 B-matrix lane order:
//   chunk ch (16 gaussians), lane l = col (l<16) -> {f0, f1, f4, f5}
//                            lane l = col + 16   -> {f2, f3, 0,  0 }
// so the raster kernel needs ONE global_load_b128 per lane per chunk.
//   packedB: [G_N/16][32] float4   (32 KB)
//   aux:     [G_N] float4(op, r, g, b) depth-sorted (16 KB)
// ---------------------------------------------------------------------------
__global__ __launch_bounds__(G_N) void gs_prep_kernel(
    const float* __restrict__ means2d,    // [G,2]
    const float* __restrict__ conics,     // [G,3]
    const float* __restrict__ colors,     // [G,3]
    const float* __restrict__ opacities,  // [G]
    const float* __restrict__ depths,     // [G]
    float4* __restrict__ packedB,         // [G/16][32]
    float4* __restrict__ auxWs)           // [G]
{
    __shared__ float sDepth[G_N];
    const int g = threadIdx.x;
    const float d = depths[g];
    sDepth[g] = d;
    __syncthreads();

    // stable ascending rank == jnp.argsort position
    int rank = 0;
    #pragma unroll 8
    for (int j = 0; j < G_N; ++j) {
        const float dj = sDepth[j];
        rank += (dj < d) || ((dj == d) && (j < g));
    }

    const float mx = means2d[2 * g + 0] - CSHIFT;
    const float my = means2d[2 * g + 1] - CSHIFT;
    const float A  = conics[3 * g + 0];
    const float Bb = conics[3 * g + 1];
    const float Cc = conics[3 * g + 2];

    // sigma = 0.5A dx^2 + B dx dy + 0.5C dy^2 as dot([px^2,px*py,py^2,px,py,1], f)
    const float f0 = 0.5f * A;
    const float f1 = Bb;
    const float f2 = 0.5f * Cc;
    const float f3 = -(A * mx + Bb * my);
    const float f4 = -(Cc * my + Bb * mx);
    const float f5 = 0.5f * A * mx * mx + 0.5f * Cc * my * my + Bb * mx * my;

    const int ch  = rank >> 4;
    const int col = rank & 15;
    packedB[ch * 32 + col]      = make_float4(f0, f1, f4, f5);  // K=0,1 | 4,5
    packedB[ch * 32 + 16 + col] = make_float4(f2, f3, 0.f, 0.f);// K=2,3 | 6,7

    auxWs[rank] = make_float4(opacities[g],
                              colors[3 * g + 0],
                              colors[3 * g + 1],
                              colors[3 * g + 2]);
}

// ---------------------------------------------------------------------------
// Kernel 2: rasterize. 8 waves/block, each wave owns 16 pixels.
// Per chunk of 16 depth-sorted gaussians:
//   sigma[16x16] = A_pix[16x8] x B_gauss[8x16] via 2x V_WMMA_F32_16X16X4_F32
//   -> LDS (stride 20, conflict-free, 16B-aligned rows)
//   -> all 32 lanes composite (upper half duplicates pixels) branchlessly
//   -> wave ballot early-exit when all 16 pixels hit FRONT_K contributors.
// ---------------------------------------------------------------------------
__global__ __launch_bounds__(256) void gs_raster_kernel(
    const float4* __restrict__ packedB,  // [G/16][32] WMMA-lane-packed features
    const float4* __restrict__ aux,      // [G] (op, r, g, b) depth-sorted
    float* __restrict__ out)             // [H*W*3]
{
    __shared__ float sSig[8][16 * SIG_STRIDE];

    const int tid  = threadIdx.x;
    const int wave = tid >> 5;
    const int lane = tid & 31;
    const int col  = lane & 15;  // A-matrix M / composite pixel row
    const int half = lane >> 4;
    const int wavePixBase = blockIdx.x * 128 + wave * 16;

    // -------- A matrix: per-pixel features (fixed for the whole kernel) ----
    // A 16x4 layout: lanes 0-15 & 16-31 both M=0..15; VGPR0 = K (0|2),
    // VGPR1 = K (1|3). Chunk1 uses K+4 (rows 6,7 are zero).
    const int  pix = wavePixBase + col;
    const float px = (float)(pix % IMG_W) + 0.5f - CSHIFT;
    const float py = (float)(pix / IMG_W) + 0.5f - CSHIFT;
    const float pxx = px * px, pxy = px * py, pyy = py * py;

    v2f a0, a1;
    a0.x = half ? pyy : pxx;   // K = 2 | 0
    a0.y = half ? px  : pxy;   // K = 3 | 1
    a1.x = half ? 0.f : py;    // K = 6 | 4
    a1.y = half ? 0.f : 1.0f;  // K = 7 | 5

    float T = 1.0f, accR = 0.0f, accG = 0.0f, accB = 0.0f;
    int cnt = 0;

    float* const sigRowW = &sSig[wave][0];                  // scatter base
    const float* const sigRowR = &sSig[wave][col * SIG_STRIDE];  // gather base

    for (int ch = 0; ch < (G_N / 16); ++ch) {
        const int gb = ch * 16;

        // one b128 load: B-matrix VGPR pair for both K-chunks, pre-packed
        const float4 bv = packedB[ch * 32 + lane];
        v2f b0, b1;
        b0.x = bv.x; b0.y = bv.y;   // K = 0,1 | 2,3
        b1.x = bv.z; b1.y = bv.w;   // K = 4,5 | 6,7 (zeros in upper half)

        v8f c = {};
        c = __builtin_amdgcn_wmma_f32_16x16x4_f32(false, a0, false, b0,
                                                  (short)0, c, false, false);
        c = __builtin_amdgcn_wmma_f32_16x16x4_f32(false, a1, false, b1,
                                                  (short)0, c, false, false);

        // C/D layout: VGPR r -> M = r (lanes 0-15) | r+8 (lanes 16-31), N = col
        #pragma unroll
        for (int r = 0; r < 8; ++r)
            sigRowW[(r + half * 8) * SIG_STRIDE + col] = c[r];

        // wave-local LDS visibility (same-wave DS ops are in order; drain them)
        asm volatile("s_wait_dscnt 0" ::: "memory");

        // gather this pixel's 16 sigmas as 4x b128 LDS loads (regs thereafter)
        float sig[16];
        #pragma unroll
        for (int q = 0; q < 4; ++q)
            *(float4*)&sig[q * 4] = *(const float4*)&sigRowR[q * 4];

        // branchless front-to-back compositing with FRONT_K counting.
        // aux[gb+n] is wave-uniform -> scalar loads.
        #pragma unroll
        for (int n = 0; n < 16; ++n) {
            const float4 ax = aux[gb + n];
            const float alpha = fminf(ax.x * __expf(-sig[n]), ALPHA_CLAMP);
            const bool ok = (sig[n] >= 0.0f) & (alpha >= ALPHA_THRESH) &
                            (cnt < FRONT_K);
            const float a_eff = ok ? alpha : 0.0f;
            const float w = T * a_eff;
            accR = fmaf(w, ax.y, accR);
            accG = fmaf(w, ax.z, accG);
            accB = fmaf(w, ax.w, accB);
            T *= (1.0f - a_eff);
            cnt += (int)ok;
        }

        // uniform early-exit: all 16 pixels of this wave saturated at FRONT_K
        if (__builtin_amdgcn_ballot_w32(cnt >= FRONT_K) == 0xFFFFFFFFu)
            break;
    }

    if (lane < 16) {
        const int p = wavePixBase + lane;
        out[p * 3 + 0] = accR;   // BG is black: img = sum(w * color)
        out[p * 3 + 1] = accG;
        out[p * 3 + 2] = accB;
    }
}

extern "C" void kernel_launch(void* const* d_in, const int* in_sizes, int n_in,
                              void* d_out, int out_size, void* d_ws, size_t ws_size,
                              hipStream_t stream) {
    const float* means2d   = (const float*)d_in[0];
    const float* conics    = (const float*)d_in[1];
    const float* colors    = (const float*)d_in[2];
    const float* opacities = (const float*)d_in[3];
    const float* depths    = (const float*)d_in[4];

    float4* packedB = (float4*)d_ws;                          // 32 KB
    float4* auxWs   = (float4*)((char*)d_ws + (G_N / 16) * 32 * sizeof(float4));
    float*  out     = (float*)d_out;

    gs_prep_kernel<<<1, G_N, 0, stream>>>(means2d, conics, colors, opacities,
                                          depths, packedB, auxWs);
    gs_raster_kernel<<<NPIX / 128, 256, 0, stream>>>(packedB, auxWs, out);
}